// KMeans_76278619177042
// MI455X (gfx1250) — compile-verified
//
#include <hip/hip_runtime.h>
#include <hip/hip_bf16.h>
#include <float.h>

#define N_POINTS 131072
#define N_CLUST  1024
#define DFEAT    256
#define ROWS_PER_BLOCK 64     // rows of x per workgroup (4 waves x 16)
#define NTILE    32           // clusters per LDS chunk (double-buffered)
#define PITCH    260          // 256+4 pad: stride%64banks=4 (conflict-free), 1040B = 65*16 (b128 aligned)
#define BLOCK_T  128
#define B128_PER_THREAD (NTILE * DFEAT / 4 / BLOCK_T)   // 16 async b128 per thread per chunk

typedef float v2f __attribute__((ext_vector_type(2)));
typedef float v8f __attribute__((ext_vector_type(8)));
typedef int   v4i __attribute__((vector_size(4 * sizeof(int))));   // matches builtin proto

typedef __attribute__((address_space(1))) v4i* gv4i_p;   // global
typedef __attribute__((address_space(3))) v4i* lv4i_p;   // LDS

#if __has_builtin(__builtin_amdgcn_global_load_async_to_lds_b128)
#define HAS_ASYNC_LDS 1
#endif

// issue one center chunk (NTILE x 256 f32) global -> LDS
__device__ __forceinline__ void issue_chunk(const float* __restrict__ centers,
                                            int c0, float* dst, int tid)
{
    for (int i = tid; i < NTILE * (DFEAT / 4); i += BLOCK_T) {
        int r = i / (DFEAT / 4), c = (i % (DFEAT / 4)) * 4;
        const float* g = centers + (size_t)(c0 + r) * DFEAT + c;
        float* l = dst + r * PITCH + c;
#ifdef HAS_ASYNC_LDS
        __builtin_amdgcn_global_load_async_to_lds_b128((gv4i_p)g, (lv4i_p)l, 0, 0);
#else
        *(float4*)l = *(const float4*)g;
#endif
    }
}

template <int CNT>
__device__ __forceinline__ void wait_asynccnt()
{
#ifdef HAS_ASYNC_LDS
#if __has_builtin(__builtin_amdgcn_s_wait_asynccnt)
    __builtin_amdgcn_s_wait_asynccnt(CNT);
#else
    asm volatile("s_wait_asynccnt %0" ::"n"(CNT));
#endif
#endif
}

// ---------------------------------------------------------------------------
// Fused distance + argmin:  d2 = ||x||^2 + ||c||^2 - 2 x.c, min/argmin over c.
// WMMA f32 16x16x4; x tile resident in LDS; center chunks double-buffered via
// async global->LDS copies overlapped with compute.
// ---------------------------------------------------------------------------
__global__ __launch_bounds__(BLOCK_T) void kmeans_dist_argmin(
    const float* __restrict__ x, const float* __restrict__ centers,
    const float* __restrict__ c2, float* __restrict__ min_dist,
    int* __restrict__ labels, float* __restrict__ counts)
{
    __shared__ float xs[ROWS_PER_BLOCK * PITCH];   // 64 x 256 x-tile (padded)
    __shared__ float cs[2][NTILE * PITCH];         // double-buffered center chunk
    __shared__ float x2s[ROWS_PER_BLOCK];          // per-row ||x||^2

    const int tid  = threadIdx.x;
    const int lane = tid & 31;
    const int wave = tid >> 5;
    const int half = lane >> 4;    // 0: rows r, 1: rows r+8 of the C tile
    const int l16  = lane & 15;    // column within 16-wide N tile
    const size_t row0 = (size_t)blockIdx.x * ROWS_PER_BLOCK;

    // prologue: kick off chunk 0 into buffer 0 before staging x
    issue_chunk(centers, 0, &cs[0][0], tid);

    // stage x tile into LDS (float2 granularity keeps 8B alignment w/ pitch 260)
    for (int i = tid; i < ROWS_PER_BLOCK * (DFEAT / 2); i += BLOCK_T) {
        int r = i / (DFEAT / 2), c = (i % (DFEAT / 2)) * 2;
        float2 v = *(const float2*)(x + (row0 + r) * DFEAT + c);
        *(float2*)(&xs[r * PITCH + c]) = v;
    }
    __syncthreads();
    for (int r = tid; r < ROWS_PER_BLOCK; r += BLOCK_T) {
        float s = 0.f;
        for (int c = 0; c < DFEAT; ++c) { float v = xs[r * PITCH + c]; s += v * v; }
        x2s[r] = s;
    }

    const int m0 = wave * 16;           // this wave's 16-row M tile
    float rmin[8];
    int   ridx[8];
#pragma unroll
    for (int r = 0; r < 8; ++r) { rmin[r] = FLT_MAX; ridx[r] = 0; }

    for (int c0 = 0; c0 < N_CLUST; c0 += NTILE) {
        const int buf = (c0 / NTILE) & 1;
        __syncthreads();   // everyone done reading cs[buf^1] -> safe to overwrite
        if (c0 + NTILE < N_CLUST) {
            issue_chunk(centers, c0 + NTILE, &cs[buf ^ 1][0], tid);
            wait_asynccnt<B128_PER_THREAD>();  // prev batch (this buf) landed; new batch in flight
        } else {
            wait_asynccnt<0>();                // tail: drain everything
        }
        __syncthreads();   // all waves' async data for cs[buf] visible

        v8f acc[2] = {};   // 2 N-subtiles of 16 clusters each

        // A frag (16x4 f32): lanes 0-15 hold K=k0,k0+1; lanes 16-31 hold K=k0+2,k0+3
        const float* ax = &xs[(m0 + l16) * PITCH + half * 2];
        const float* bx = &cs[buf][l16 * PITCH + half * 2];
#pragma unroll 4
        for (int k0 = 0; k0 < DFEAT; k0 += 4) {
            v2f a = *(const v2f*)(ax + k0);
#pragma unroll
            for (int t = 0; t < 2; ++t) {
                v2f b = *(const v2f*)(bx + t * 16 * PITCH + k0);
                acc[t] = __builtin_amdgcn_wmma_f32_16x16x4_f32(
                    /*neg_a=*/false, a, /*neg_b=*/false, b,
                    /*c_mod=*/(short)0, acc[t], /*reuse_a=*/false, /*reuse_b=*/false);
            }
        }

        // fold: val = ||c_n||^2 - 2 * dot   (||x||^2 is row-constant, added at end)
#pragma unroll
        for (int t = 0; t < 2; ++t) {
            int n = c0 + t * 16 + l16;
            float cn = c2[n];
#pragma unroll
            for (int r = 0; r < 8; ++r) {
                float val = cn - 2.0f * acc[t][r];
                if (val < rmin[r]) { rmin[r] = val; ridx[r] = n; }
            }
        }
    }

    // cross-lane min within each 16-lane half (lanes share the same 8 rows per half)
#pragma unroll
    for (int r = 0; r < 8; ++r) {
#pragma unroll
        for (int m = 8; m >= 1; m >>= 1) {
            float ov = __shfl_xor(rmin[r], m, 32);
            int   oi = __shfl_xor(ridx[r], m, 32);
            if (ov < rmin[r] || (ov == rmin[r] && oi < ridx[r])) { rmin[r] = ov; ridx[r] = oi; }
        }
    }

    if (l16 == 0) {
#pragma unroll
        for (int r = 0; r < 8; ++r) {
            int lrow = m0 + r + 8 * half;   // C layout: VGPR r -> rows r (half 0) / r+8 (half 1)
            size_t grow = row0 + lrow;
            float d2 = x2s[lrow] + rmin[r];
            min_dist[grow] = sqrtf(fmaxf(d2, 0.f));
            labels[grow]   = ridx[r];
            atomicAdd(&counts[ridx[r]], 1.0f);
        }
    }
}

// ---------------------------------------------------------------------------
__global__ void center_norms(const float* __restrict__ centers, float* __restrict__ c2)
{
    int k = blockIdx.x * blockDim.x + threadIdx.x;
    if (k < N_CLUST) {
        const float2* p = (const float2*)(centers + (size_t)k * DFEAT);
        float s = 0.f;
        for (int i = 0; i < DFEAT / 2; ++i) { float2 v = p[i]; s += v.x * v.x + v.y * v.y; }
        c2[k] = s;
    }
}

__global__ void zero_accum(float* __restrict__ sums, float* __restrict__ counts)
{
    int i = blockIdx.x * blockDim.x + threadIdx.x;
    if (i < N_CLUST * DFEAT) sums[i] = 0.f;
    else if (i < N_CLUST * DFEAT + N_CLUST) counts[i - N_CLUST * DFEAT] = 0.f;
}

// one thread per (point, 4-feature chunk); consecutive threads -> consecutive features
__global__ __launch_bounds__(256) void scatter_sums(
    const float* __restrict__ x, const int* __restrict__ labels, float* __restrict__ sums)
{
    size_t idx = (size_t)blockIdx.x * blockDim.x + threadIdx.x;
    int p = (int)(idx >> 6);
    int c = (int)(idx & 63) << 2;
    int lab = labels[p];
    float4 v = *(const float4*)(x + (size_t)p * DFEAT + c);
    float* dst = sums + (size_t)lab * DFEAT + c;
    atomicAdd(dst + 0, v.x);
    atomicAdd(dst + 1, v.y);
    atomicAdd(dst + 2, v.z);
    atomicAdd(dst + 3, v.w);
}

__global__ void finalize_centers(const float* __restrict__ centers,
                                 const float* __restrict__ counts,
                                 float* __restrict__ newc)
{
    int idx = blockIdx.x * blockDim.x + threadIdx.x;
    if (idx < N_CLUST * DFEAT) {
        int k = idx >> 8;    // DFEAT == 256
        float cnt = counts[k];
        newc[idx] = (cnt > 0.5f) ? (newc[idx] / cnt) : centers[idx];
    }
}

// ---------------------------------------------------------------------------
extern "C" void kernel_launch(void* const* d_in, const int* in_sizes, int n_in,
                              void* d_out, int out_size, void* d_ws, size_t ws_size,
                              hipStream_t stream)
{
    const float* x       = (const float*)d_in[0];   // [131072, 256] f32
    const float* centers = (const float*)d_in[1];   // [1024, 256]  f32

    float* out      = (float*)d_out;
    float* min_dist = out;                          // [131072] f32
    int*   labels   = (int*)(out + N_POINTS);       // [131072] i32
    float* newc     = out + 2 * N_POINTS;           // [1024, 256] f32 (doubles as sums accum)

    float* counts = (float*)d_ws;                   // [1024] f32
    float* c2     = counts + N_CLUST;               // [1024] f32

    (void)in_sizes; (void)n_in; (void)out_size; (void)ws_size;

    zero_accum<<<(N_CLUST * DFEAT + N_CLUST + 255) / 256, 256, 0, stream>>>(newc, counts);
    center_norms<<<(N_CLUST + 255) / 256, 256, 0, stream>>>(centers, c2);
    kmeans_dist_argmin<<<N_POINTS / ROWS_PER_BLOCK, BLOCK_T, 0, stream>>>(
        x, centers, c2, min_dist, labels, counts);
    scatter_sums<<<(size_t)N_POINTS * 64 / 256, 256, 0, stream>>>(x, labels, newc);
    finalize_centers<<<(N_CLUST * DFEAT + 255) / 256, 256, 0, stream>>>(centers, counts, newc);
}